// ConvertMomentsTEtoParamsTEGaussLayerFV_25237227831465
// MI455X (gfx1250) — compile-verified
//
#include <hip/hip_runtime.h>

typedef __attribute__((ext_vector_type(2))) float v2f;
typedef __attribute__((ext_vector_type(4))) float v4f;
typedef __attribute__((ext_vector_type(8))) float v8f;

#define NV 512            // matrix dimension
#define BM 128            // block tile M
#define BN 64             // block tile N
#define KB 32             // K panel per LDS stage
#define APAD (KB + 2)     // A row stride (floats)
#define BSTRIDE 160       // B pair-row stride: 64*2 + 32 pad (bank-shift by 32)

// C = (NEGATE ? -1 : 1) * opA(A) @ maskB(B)
// opA(A)   = TRANS_A ? A^T : A
// maskB(B) = MASK_B ? tril(B,-1) + 0.5*diag(B) : B
template <bool TRANS_A, bool MASK_B, bool NEGATE>
__global__ __launch_bounds__(256) void gemm512_wmma(
    const float* __restrict__ A, const float* __restrict__ Bm,
    float* __restrict__ C) {
  __shared__ float As[BM][APAD];            // As[m][k]
  __shared__ float Bp[(KB / 2) * BSTRIDE];  // K-pair interleaved: [kp][n][2]

  const int tid = threadIdx.x;          // 256 threads = 8 waves
  const int colBase = blockIdx.x * BN;  // N tile base
  const int rowBase = blockIdx.y * BM;  // M tile base
  const size_t boff = (size_t)blockIdx.z * NV * NV;
  A += boff;
  Bm += boff;
  C += boff;

  const int wave = tid >> 5;
  const int lane = tid & 31;
  const int waveM = wave & 3;   // 4 wave-rows of 32
  const int waveN = wave >> 2;  // 2 wave-cols of 32
  const int laneLo = lane & 15;
  const int laneHi = lane >> 4;  // selects K pair (and M upper half on store)

  const int m0 = waveM * 32;            // wave's M base inside block
  const int bc0 = waveN * 32 + laneLo;  // wave's N lane-col inside block

  v8f acc00 = {}, acc01 = {}, acc10 = {}, acc11 = {};

  for (int kBase = 0; kBase < NV; kBase += KB) {
    // ---- stage A panel (128 x 32) ----
    if (!TRANS_A) {
#pragma unroll
      for (int i = 0; i < 4; ++i) {
        int idx = tid + i * 256;  // 0..1023 float4s
        int c4 = idx & 7;         // 8 float4 per row of 32
        int r = idx >> 3;         // 0..127
        v4f v = *(const v4f*)&A[(size_t)(rowBase + r) * NV + kBase + c4 * 4];
        v2f lo = {v.x, v.y}, hi = {v.z, v.w};
        *(v2f*)&As[r][c4 * 4] = lo;      // (r*APAD + 4c) even -> 8B aligned
        *(v2f*)&As[r][c4 * 4 + 2] = hi;
      }
    } else {
#pragma unroll
      for (int i = 0; i < 4; ++i) {
        int idx = tid + i * 256;
        int m4 = idx & 31;  // 32 float4 covering 128 m (coalesced in m)
        int k = idx >> 5;   // 0..31
        v4f v = *(const v4f*)&A[(size_t)(kBase + k) * NV + rowBase + m4 * 4];
        As[m4 * 4 + 0][k] = v.x;
        As[m4 * 4 + 1][k] = v.y;
        As[m4 * 4 + 2][k] = v.z;
        As[m4 * 4 + 3][k] = v.w;
      }
    }
    // ---- stage B panel (32 x 64), K-pair interleaved ----
#pragma unroll
    for (int i = 0; i < 2; ++i) {
      int idx = tid + i * 256;  // 0..511 float4s
      int n4 = idx & 15;        // 16 float4 per row of 64
      int k = idx >> 4;         // 0..31
      v4f v = *(const v4f*)&Bm[(size_t)(kBase + k) * NV + colBase + n4 * 4];
      const int kp = k >> 1;
      const int ko = k & 1;
#pragma unroll
      for (int j = 0; j < 4; ++j) {
        float val = v[j];
        if (MASK_B) {
          int gk = kBase + k;
          int gn = colBase + n4 * 4 + j;
          val = (gk > gn) ? val : ((gk == gn) ? 0.5f * val : 0.0f);
        }
        Bp[kp * BSTRIDE + (n4 * 4 + j) * 2 + ko] = val;
      }
    }
    __syncthreads();

    // ---- 8 WMMA K-steps of 4; 4 WMMAs per step (2 A-frags x 2 B-frags) ----
#pragma unroll
    for (int ks = 0; ks < KB; ks += 4) {
      const int kOff = ks + 2 * laneHi;       // even -> 8B-aligned
      const int kp = (ks >> 1) + laneHi;      // pair row in Bp
      v2f a0 = *(const v2f*)&As[m0 + laneLo][kOff];
      v2f a1 = *(const v2f*)&As[m0 + 16 + laneLo][kOff];
      v2f b0 = *(const v2f*)&Bp[kp * BSTRIDE + bc0 * 2];         // one b64
      v2f b1 = *(const v2f*)&Bp[kp * BSTRIDE + (bc0 + 16) * 2];  // one b64
      acc00 = __builtin_amdgcn_wmma_f32_16x16x4_f32(false, a0, false, b0,
                                                    (short)0, acc00, false, false);
      acc01 = __builtin_amdgcn_wmma_f32_16x16x4_f32(false, a0, false, b1,
                                                    (short)0, acc01, false, false);
      acc10 = __builtin_amdgcn_wmma_f32_16x16x4_f32(false, a1, false, b0,
                                                    (short)0, acc10, false, false);
      acc11 = __builtin_amdgcn_wmma_f32_16x16x4_f32(false, a1, false, b1,
                                                    (short)0, acc11, false, false);
    }
    __syncthreads();
  }

  // ---- store: VGPR r of a 16x16 f32 D -> rows (r, r+8), col = lane&15 ----
  const int gr = rowBase + m0 + laneHi * 8;
  const int gc = colBase + waveN * 32 + laneLo;
#pragma unroll
  for (int r = 0; r < 8; ++r) {
    float v00 = acc00[r], v01 = acc01[r], v10 = acc10[r], v11 = acc11[r];
    if (NEGATE) {
      v00 = -v00;
      v01 = -v01;
      v10 = -v10;
      v11 = -v11;
    }
    C[(size_t)(gr + r) * NV + gc] = v00;
    C[(size_t)(gr + r) * NV + gc + 16] = v01;
    C[(size_t)(gr + 16 + r) * NV + gc] = v10;
    C[(size_t)(gr + 16 + r) * NV + gc + 16] = v11;
  }
}

__global__ void copy_f32(const float* __restrict__ src, float* __restrict__ dst,
                         int n) {
  int i = blockIdx.x * blockDim.x + threadIdx.x;
  if (i < n) dst[i] = src[i];
}

extern "C" void kernel_launch(void* const* d_in, const int* in_sizes, int n_in,
                              void* d_out, int out_size, void* d_ws,
                              size_t ws_size, hipStream_t stream) {
  const float* muTE = (const float*)d_in[0];   // [B, 512]
  const float* covTE = (const float*)d_in[1];  // [B, 512, 512]
  const float* chol = (const float*)d_in[2];   // [B, 512, 512]

  const int nMu = in_sizes[0];  // B*512
  const int batch = nMu / NV;   // 128

  float* out = (float*)d_out;
  float* outMu = out;          // output 0: muTE passthrough
  float* outChol = out + nMu;  // output 1: cholTE [B,512,512]

  float* T1 = outChol;         // stage covTE@chol in the cholTE slot
  float* phi = (float*)d_ws;   // needs B*512*512*4 = 128 MB scratch

  // 1) muTE passthrough
  copy_f32<<<(nMu + 255) / 256, 256, 0, stream>>>(muTE, outMu, nMu);

  dim3 grid(NV / BN, NV / BM, batch);  // (8, 4, 128)
  dim3 block(256);

  // 2) T1 = covTE @ chol
  gemm512_wmma<false, false, false><<<grid, block, 0, stream>>>(covTE, chol, T1);
  // 3) phi = chol^T @ T1
  gemm512_wmma<true, false, false><<<grid, block, 0, stream>>>(chol, T1, phi);
  // 4) cholTE = -(chol @ mask(phi))   (overwrites T1 slot; T1 never read again)
  gemm512_wmma<false, true, true><<<grid, block, 0, stream>>>(chol, phi, outChol);
}